// GAT_55018531062471
// MI455X (gfx1250) — compile-verified
//
#include <hip/hip_runtime.h>
#include <hip/hip_bf16.h>

// ---------------------------------------------------------------------------
// GAT (2x GATConv + Linear) for MI455X / gfx1250, wave32, WMMA f32 16x16x4.
// ---------------------------------------------------------------------------

typedef float v2f __attribute__((ext_vector_type(2)));
typedef float v8f __attribute__((ext_vector_type(8)));

// ---------------- float atomic max via integer atomics ----------------------
__device__ __forceinline__ void atomicMaxFloat(float* addr, float val) {
    if (val >= 0.0f) {
        atomicMax((int*)addr, __float_as_int(val));
    } else {
        atomicMin((unsigned int*)addr, __float_as_uint(val));
    }
}

// ---------------- WMMA GEMM: C[M x NREAL] = A[M x K] * B[K x NREAL] (+bias) --
// One block = NTILES waves; block handles a 16-row stripe, wave w handles
// columns [16w, 16w+16). Both operands staged in LDS so the K-loop is pure
// ds_load_b64 + v_wmma with no branches:
//   A: 16 consecutive rows, contiguous copy.           lane: M=lane%16,
//      K = k0 + 2*(lane/16) + {0,1}  -> one b64 load.
//   B: zero-padded to NPAD cols, K-pair interleaved:
//      lB[((k/2)*NPAD + col)*2 + (k%2)] = B[k][col]    -> one b64 load.
template<int K, int NTILES, int NREAL>
__global__ __launch_bounds__(32 * NTILES)
void wmma_gemm(const float* __restrict__ A, const float* __restrict__ B,
               const float* __restrict__ bias, float* __restrict__ C, int M) {
    constexpr int NPAD = NTILES * 16;
    __shared__ float lA[16 * K];
    __shared__ float lB[K * NPAD];
    const int tid  = threadIdx.x;
    const int lane = tid & 31;
    const int wave = tid >> 5;           // N-tile index
    const int m0   = blockIdx.x * 16;

    // cooperative, fully-coalesced copy of 16 consecutive rows of A
    const float* arow = A + (size_t)m0 * K;
    for (int i = tid; i < 16 * K; i += 32 * NTILES) {
        const int r = m0 + i / K;
        lA[i] = (r < M) ? arow[i] : 0.0f;
    }
    // stage B zero-padded + pair-interleaved
    for (int i = tid; i < K * NPAD; i += 32 * NTILES) {
        const int k = i / NPAD, col = i % NPAD;
        const float v = (col < NREAL) ? B[(size_t)k * NREAL + col] : 0.0f;
        lB[((k >> 1) * NPAD + col) * 2 + (k & 1)] = v;
    }
    __syncthreads();

    const int mloc  = lane & 15;         // M index for A operand
    const int khalf = lane >> 4;         // 0 or 1
    const int col   = wave * 16 + (lane & 15);

    const float* aBase = &lA[(size_t)mloc * K];
    v8f acc = {};
#pragma unroll
    for (int k0 = 0; k0 < K; k0 += 4) {
        const int ka = k0 + 2 * khalf;                   // K = vgpr + 2*(lane/16)
        const v2f a = *(const v2f*)(aBase + ka);         // ds_load_b64
        const v2f b = *(const v2f*)(&lB[((ka >> 1) * NPAD + col) * 2]); // ds_load_b64
        acc = __builtin_amdgcn_wmma_f32_16x16x4_f32(
            /*neg_a=*/false, a, /*neg_b=*/false, b,
            /*c_mod=*/(short)0, acc, /*reuse_a=*/false, /*reuse_b=*/false);
    }

    if (NPAD == NREAL || col < NREAL) {
        const float bv = bias ? bias[col] : 0.0f;
#pragma unroll
        for (int v = 0; v < 8; ++v) {
            const int row = m0 + v + 8 * khalf;   // M = vgpr + 8*(lane/16)
            if (row < M) C[(size_t)row * NREAL + col] = acc[v] + bv;
        }
    }
}

// ---------------- per-(node,head) attention coefficients --------------------
__global__ void attn_coef(const float* __restrict__ h,
                          const float* __restrict__ a_src,
                          const float* __restrict__ a_dst,
                          float* __restrict__ as, float* __restrict__ ad,
                          int n, int H, int C) {
    const int idx = blockIdx.x * blockDim.x + threadIdx.x;
    if (idx >= n * H) return;
    const int node = idx / H, hh = idx % H;
    const float* hp = h + (size_t)node * H * C + hh * C;
    float s = 0.0f, d = 0.0f;
    for (int c = 0; c < C; ++c) {
        const float v = hp[c];
        s += v * a_src[hh * C + c];
        d += v * a_dst[hh * C + c];
    }
    as[idx] = s;
    ad[idx] = d;
}

// ---------------- init: m=-inf, denom=0, accum=0 ----------------------------
__global__ void init_seg(float* __restrict__ m, float* __restrict__ denom,
                         float* __restrict__ accum, int nH, int nF) {
    const int idx = blockIdx.x * blockDim.x + threadIdx.x;
    if (idx < nF) accum[idx] = 0.0f;
    if (idx < nH) { m[idx] = -__builtin_inff(); denom[idx] = 0.0f; }
}

// ---------------- edge pass 1: leaky-relu + segment max ---------------------
__global__ void edge_max(const int* __restrict__ ei, int E, int n,
                         const float* __restrict__ as, const float* __restrict__ ad,
                         int H, float* __restrict__ m) {
    const int e = blockIdx.x * blockDim.x + threadIdx.x;
    const int ET = E + n;
    if (e >= ET) return;
    int src, dst;
    if (e < E) { src = ei[e]; dst = ei[E + e]; } else { src = dst = e - E; }
    for (int h = 0; h < H; ++h) {
        float v = as[src * H + h] + ad[dst * H + h];
        v = v > 0.0f ? v : 0.2f * v;               // leaky_relu(0.2)
        atomicMaxFloat(&m[dst * H + h], v);
    }
}

// ---------------- edge pass 2: ex = exp(e - m[dst]); denom += ex ------------
__global__ void edge_expdenom(const int* __restrict__ ei, int E, int n,
                              const float* __restrict__ as, const float* __restrict__ ad,
                              const float* __restrict__ m, int H,
                              float* __restrict__ exbuf, float* __restrict__ denom) {
    const int e = blockIdx.x * blockDim.x + threadIdx.x;
    const int ET = E + n;
    if (e >= ET) return;
    int src, dst;
    if (e < E) { src = ei[e]; dst = ei[E + e]; } else { src = dst = e - E; }
    for (int h = 0; h < H; ++h) {
        float v = as[src * H + h] + ad[dst * H + h];
        v = v > 0.0f ? v : 0.2f * v;
        const float ex = expf(v - m[dst * H + h]);
        exbuf[(size_t)e * H + h] = ex;
        atomicAdd(&denom[dst * H + h], ex);
    }
}

// ---------------- edge pass 3: accum[dst] += ex * h[src] --------------------
// One thread per (edge, feature). Division by denom deferred to norm_elu.
template<int H, int C>
__global__ __launch_bounds__(256)
void edge_scatter(const int* __restrict__ ei, int E, int n,
                  const float* __restrict__ hfeat,
                  const float* __restrict__ exbuf,
                  float* __restrict__ accum) {
    constexpr int F = H * C;
    constexpr int EPB = 256 / F;
    const int tid = threadIdx.x;
    const int e = blockIdx.x * EPB + tid / F;
    const int f = tid % F;
    const int ET = E + n;
    if (e >= ET) return;
    int src, dst;
    if (e < E) { src = ei[e]; dst = ei[E + e]; } else { src = dst = e - E; }
    const int h = f / C;
    const float ex = exbuf[(size_t)e * H + h];
    const float val = ex * hfeat[(size_t)src * F + f];
    atomicAdd(&accum[(size_t)dst * F + f], val);
}

// ---------------- normalize + bias + ELU ------------------------------------
__global__ void norm_elu(const float* __restrict__ accum,
                         const float* __restrict__ denom,
                         const float* __restrict__ bias,
                         float* __restrict__ out, int n, int H, int C) {
    const int idx = blockIdx.x * blockDim.x + threadIdx.x;
    const int F = H * C;
    if (idx >= n * F) return;
    const int node = idx / F, f = idx % F, h = f / C;
    float v = accum[idx] / (denom[node * H + h] + 1e-16f) + bias[f];
    out[idx] = v > 0.0f ? v : (expf(v) - 1.0f);
}

// ---------------------------------------------------------------------------
extern "C" void kernel_launch(void* const* d_in, const int* in_sizes, int n_in,
                              void* d_out, int out_size, void* d_ws, size_t ws_size,
                              hipStream_t stream) {
    constexpr int IN = 128, HID = 32, HEADS = 2, OUTD = 40;
    constexpr int F1 = HEADS * HID;     // 64

    const float* x       = (const float*)d_in[0];
    const int*   ei      = (const int*)d_in[1];
    const float* W1      = (const float*)d_in[2];
    const float* a_src1  = (const float*)d_in[3];
    const float* a_dst1  = (const float*)d_in[4];
    const float* b1      = (const float*)d_in[5];
    const float* W2      = (const float*)d_in[6];
    const float* a_src2  = (const float*)d_in[7];
    const float* a_dst2  = (const float*)d_in[8];
    const float* b2      = (const float*)d_in[9];
    const float* Wc      = (const float*)d_in[10];
    const float* bc      = (const float*)d_in[11];
    float* out = (float*)d_out;

    const int N  = in_sizes[0] / IN;     // 50000
    const int E  = in_sizes[1] / 2;      // 800000
    const int ET = E + N;                // edges incl. self-loops

    // ---- bump-allocate workspace ----
    char* wsp = (char*)d_ws;
    auto alloc = [&](size_t bytes) -> float* {
        float* p = (float*)wsp;
        wsp += (bytes + 255) & ~(size_t)255;
        return p;
    };
    float* h1     = alloc((size_t)N * F1 * 4);
    float* accum1 = alloc((size_t)N * F1 * 4);
    float* t1     = alloc((size_t)N * F1 * 4);
    float* h2     = alloc((size_t)N * HID * 4);
    float* accum2 = alloc((size_t)N * HID * 4);
    float* t2     = alloc((size_t)N * HID * 4);
    float* as1    = alloc((size_t)N * HEADS * 4);
    float* ad1    = alloc((size_t)N * HEADS * 4);
    float* m1     = alloc((size_t)N * HEADS * 4);
    float* den1   = alloc((size_t)N * HEADS * 4);
    float* as2    = alloc((size_t)N * 4);
    float* ad2    = alloc((size_t)N * 4);
    float* m2     = alloc((size_t)N * 4);
    float* den2   = alloc((size_t)N * 4);
    float* exbuf  = alloc((size_t)ET * HEADS * 4);
    (void)ws_size; (void)n_in; (void)out_size;

    const int TPB = 256;
    const int Mtiles = (N + 15) / 16;

    // ================= Layer 1: GATConv(128 -> 32, heads=2, concat) =========
    init_seg<<<(N * F1 + TPB - 1) / TPB, TPB, 0, stream>>>(m1, den1, accum1,
                                                           N * HEADS, N * F1);
    wmma_gemm<IN, F1 / 16, F1><<<Mtiles, 32 * (F1 / 16), 0, stream>>>(
        x, W1, nullptr, h1, N);
    attn_coef<<<(N * HEADS + TPB - 1) / TPB, TPB, 0, stream>>>(
        h1, a_src1, a_dst1, as1, ad1, N, HEADS, HID);
    edge_max<<<(ET + TPB - 1) / TPB, TPB, 0, stream>>>(ei, E, N, as1, ad1, HEADS, m1);
    edge_expdenom<<<(ET + TPB - 1) / TPB, TPB, 0, stream>>>(
        ei, E, N, as1, ad1, m1, HEADS, exbuf, den1);
    {
        constexpr int EPB = 256 / (HEADS * HID);   // 4 edges / block
        edge_scatter<HEADS, HID><<<(ET + EPB - 1) / EPB, 256, 0, stream>>>(
            ei, E, N, h1, exbuf, accum1);
    }
    norm_elu<<<(N * F1 + TPB - 1) / TPB, TPB, 0, stream>>>(
        accum1, den1, b1, t1, N, HEADS, HID);

    // ================= Layer 2: GATConv(64 -> 32, heads=1) ==================
    init_seg<<<(N * HID + TPB - 1) / TPB, TPB, 0, stream>>>(m2, den2, accum2,
                                                            N, N * HID);
    wmma_gemm<F1, HID / 16, HID><<<Mtiles, 32 * (HID / 16), 0, stream>>>(
        t1, W2, nullptr, h2, N);
    attn_coef<<<(N + TPB - 1) / TPB, TPB, 0, stream>>>(
        h2, a_src2, a_dst2, as2, ad2, N, 1, HID);
    edge_max<<<(ET + TPB - 1) / TPB, TPB, 0, stream>>>(ei, E, N, as2, ad2, 1, m2);
    edge_expdenom<<<(ET + TPB - 1) / TPB, TPB, 0, stream>>>(
        ei, E, N, as2, ad2, m2, 1, exbuf, den2);
    {
        constexpr int EPB = 256 / HID;             // 8 edges / block
        edge_scatter<1, HID><<<(ET + EPB - 1) / EPB, 256, 0, stream>>>(
            ei, E, N, h2, exbuf, accum2);
    }
    norm_elu<<<(N * HID + TPB - 1) / TPB, TPB, 0, stream>>>(
        accum2, den2, b2, t2, N, 1, HID);

    // ================= Classifier: Linear(32 -> 40) =========================
    wmma_gemm<HID, 3, OUTD><<<Mtiles, 32 * 3, 0, stream>>>(t2, Wc, bc, out, N);
}